// QuantumAttnBlk_72602127171980
// MI455X (gfx1250) — compile-verified
//
#include <hip/hip_runtime.h>

typedef __attribute__((ext_vector_type(2))) float v2f;
typedef __attribute__((ext_vector_type(8))) float v8f;

// ---------------------------------------------------------------------------
// Stage 1: precompute the 4x12 coefficient matrix (from `weights`) into the
// WMMA A-operand layout for V_WMMA_F32_16X16X4_F32, split into 3 K=4 chunks.
// A layout (16x4 f32): lanes 0-15 hold rows M=0..15 with VGPR0=K0, VGPR1=K1;
// lanes 16-31 hold the same rows with VGPR0=K2, VGPR1=K3.
// Coefficients (Heisenberg picture, U3(th,phi,lam)^† Z U3 = aX+bY+cZ):
//   alpha_w = -sin(th)cos(lam), beta_w = sin(th)sin(lam), gamma_w = cos(th)
// Global column g = 3w+{0,1,2} holds {alpha,beta,gamma}_w on row w.
// ---------------------------------------------------------------------------
__global__ void qc_prep(const float* __restrict__ wt, float* __restrict__ Abuf) {
    int L = threadIdx.x;              // 0..31, one wave
    float coef[12];
#pragma unroll
    for (int w = 0; w < 4; ++w) {
        float th = wt[3 * w + 0];
        float lam = wt[3 * w + 2];
        float st, ct, sl, cl;
        __sincosf(th, &st, &ct);
        __sincosf(lam, &sl, &cl);
        coef[3 * w + 0] = -st * cl;   // alpha_w
        coef[3 * w + 1] =  st * sl;   // beta_w
        coef[3 * w + 2] =  ct;        // gamma_w
    }
    int row = L & 15;
    bool lo = (L < 16);
#pragma unroll
    for (int c = 0; c < 3; ++c) {
        // global columns for this lane: lo -> {4c, 4c+1}, hi -> {4c+2, 4c+3}
        float axlo = (((4 * c + 0) / 3) == row) ? coef[4 * c + 0] : 0.0f;
        float aylo = (((4 * c + 1) / 3) == row) ? coef[4 * c + 1] : 0.0f;
        float axhi = (((4 * c + 2) / 3) == row) ? coef[4 * c + 2] : 0.0f;
        float ayhi = (((4 * c + 3) / 3) == row) ? coef[4 * c + 3] : 0.0f;
        float2 a;
        a.x = lo ? axlo : axhi;
        a.y = lo ? aylo : ayhi;
        ((float2*)Abuf)[c * 32 + L] = a;
    }
}

// ---------------------------------------------------------------------------
// Stage 2: fused per-element pass. Each lane owns one batch element:
//   - 12 sincos -> Bloch vectors m_q of the 4 product-state qubits
//   - 12 Pauli-string monomials t[0..11]
//   - out(4xB) = Coef(4x12) x T(12xB) via 3x V_WMMA_F32_16X16X4_F32 per
//     16-element group (two groups per wave; B operands built with
//     __shfl_xor(.,16) half-wave exchanges, no duplicated transcendentals).
// Memory: 48 B in + 16 B out per element -> bandwidth bound (~2.7us @23.3TB/s).
// ---------------------------------------------------------------------------
__global__ __launch_bounds__(256) void qc_main(const float* __restrict__ x,
                                               const float* __restrict__ Abuf,
                                               float* __restrict__ out) {
    int lane = threadIdx.x & 31;
    int wave = blockIdx.x * (blockDim.x >> 5) + (threadIdx.x >> 5);
    long long e = (long long)wave * 32 + lane;

    const float4* xp = (const float4*)(x + e * 12);
    float4 g0 = xp[0];
    float4 g1 = xp[1];
    float4 g2 = xp[2];
    float ang[12] = {g0.x, g0.y, g0.z, g0.w, g1.x, g1.y, g1.z, g1.w,
                     g2.x, g2.y, g2.z, g2.w};

    float mx[4], my[4], mz[4];
#pragma unroll
    for (int q = 0; q < 4; ++q) {
        float s1, c1, s2, c2, s3, c3;
        __sincosf(ang[3 * q + 0], &s1, &c1);
        __sincosf(ang[3 * q + 1], &s2, &c2);
        __sincosf(ang[3 * q + 2], &s3, &c3);
        float cc = c1 * c2;
        mx[q] = __builtin_fmaf(cc, c3, -(s1 * s3));
        my[q] = __builtin_fmaf(cc, s3, s1 * c3);
        mz[q] = -(c1 * s2);
    }

    // Pauli-string monomials (CNOT-ring Clifford conjugation of X/Y/Z_w):
    float z01 = mz[0] * mz[1];
    float z23 = mz[2] * mz[3];
    float x01 = mx[0] * mx[1];
    float z012 = z01 * mz[2];
    float t[12];
    t[0]  = x01;                              // X0X1           (alpha0)
    t[1]  = mx[0] * my[1] * z23;              // X0Y1Z2Z3       (beta0)
    t[2]  = mz[1] * z23;                      // Z1Z2Z3         (gamma0)
    t[3]  = mx[1] * mx[2];                    // X1X2           (alpha1)
    t[4]  = mz[0] * my[1] * mx[2];            // Z0Y1X2         (beta1)
    t[5]  = z01;                              // Z0Z1           (gamma1)
    t[6]  = mx[2] * mx[3];                    // X2X3           (alpha2)
    t[7]  = z01 * my[2] * mx[3];              // Z0Z1Y2X3       (beta2)
    t[8]  = z012;                             // Z0Z1Z2         (gamma2)
    t[9]  = x01 * mx[3];                      // X0X1X3         (alpha3)
    t[10] = -(my[0] * my[1] * mz[2] * my[3]); // -Y0Y1Z2Y3      (beta3)
    t[11] = z012 * mz[3];                     // Z0Z1Z2Z3       (gamma3)

    // Half-wave exchange so each 16-element WMMA group can see the other
    // half's monomials without recomputing the sincos chain.
    float ts[12];
#pragma unroll
    for (int k = 0; k < 12; ++k) ts[k] = __shfl_xor(t[k], 16, 32);

    bool lo = (lane < 16);
    v8f accA = {};   // elements waveBase + 0..15
    v8f accB = {};   // elements waveBase + 16..31
#pragma unroll
    for (int c = 0; c < 3; ++c) {
        float2 af = ((const float2*)Abuf)[c * 32 + lane];
        v2f a;
        a[0] = af.x;
        a[1] = af.y;
        // B (4x16): VGPR0 = rows K0 (lanes 0-15) / K2 (lanes 16-31),
        //           VGPR1 = rows K1 / K3  (dual of the documented A layout)
        v2f bA, bB;
        bA[0] = lo ? t[4 * c + 0] : ts[4 * c + 2];
        bA[1] = lo ? t[4 * c + 1] : ts[4 * c + 3];
        bB[0] = lo ? ts[4 * c + 0] : t[4 * c + 2];
        bB[1] = lo ? ts[4 * c + 1] : t[4 * c + 3];
        accA = __builtin_amdgcn_wmma_f32_16x16x4_f32(false, a, false, bA,
                                                     (short)0, accA, false, false);
        accB = __builtin_amdgcn_wmma_f32_16x16x4_f32(false, a, false, bB,
                                                     (short)0, accB, false, false);
    }

    // D layout: lanes 0-15 hold N=element, VGPR m = output wire m (m=0..3).
    if (lo) {
        float4 oA = make_float4(accA[0], accA[1], accA[2], accA[3]);
        float4 oB = make_float4(accB[0], accB[1], accB[2], accB[3]);
        float4* op = (float4*)out;
        long long base = (long long)wave * 32;
        op[base + lane] = oA;
        op[base + 16 + lane] = oB;
    }
}

extern "C" void kernel_launch(void* const* d_in, const int* in_sizes, int n_in,
                              void* d_out, int out_size, void* d_ws, size_t ws_size,
                              hipStream_t stream) {
    const float* x  = (const float*)d_in[0];   // (B, 12) f32
    const float* wt = (const float*)d_in[1];   // (1, 4, 3) f32
    float* out = (float*)d_out;                // (B, 4) f32
    float* Abuf = (float*)d_ws;                // 3 * 32 * float2 = 768 B

    int batch = in_sizes[0] / 12;              // 1048576

    qc_prep<<<1, 32, 0, stream>>>(wt, Abuf);

    // 256 threads = 8 waves = 256 elements per block; batch divisible by 256.
    int blocks = batch / 256;
    qc_main<<<blocks, 256, 0, stream>>>(x, Abuf, out);
}